// EmbeddingMul_5995774345886
// MI455X (gfx1250) — compile-verified
//
#include <hip/hip_runtime.h>
#include <hip/hip_bf16.h>

// EmbeddingMul == one_hot @ weight == row gather: out[row,:] = weight[idx[row],:]
// Pure bandwidth problem (16 MB total @ 23.3 TB/s ~ 0.7 us). Use the gfx1250
// async Global->LDS->Global path (ASYNCcnt) so data never touches VGPRs.

typedef int v4i __attribute__((__vector_size__(16)));

#define LDS_AS __attribute__((address_space(3)))

__device__ __forceinline__ unsigned lds_byte_addr(const void* p) {
    // generic -> LDS(3) addrspace cast, then raw 32-bit LDS byte offset (asm path).
    return (unsigned)(unsigned long long)(LDS_AS const char*)p;
}

__device__ __forceinline__ void async_g2lds_b128(void* lds, const void* g) {
#if __has_builtin(__builtin_amdgcn_global_load_async_to_lds_b128)
    // signature (from hipcc diagnostic): (v4i* gaddr, v4i* lds, Ii offset, Ii cpol)
    __builtin_amdgcn_global_load_async_to_lds_b128(
        (v4i*)g, (v4i*)lds, /*imm offset*/0, /*cpol*/0);
#else
    asm volatile("global_load_async_to_lds_b128 %0, %1, off"
                 :: "v"(lds_byte_addr(lds)), "v"(g) : "memory");
#endif
}

__device__ __forceinline__ void async_lds2g_b128(void* g, const void* lds) {
#if __has_builtin(__builtin_amdgcn_global_store_async_from_lds_b128)
    __builtin_amdgcn_global_store_async_from_lds_b128(
        (v4i*)g, (v4i*)lds, /*imm offset*/0, /*cpol*/0);
#else
    asm volatile("global_store_async_from_lds_b128 %0, %1, off"
                 :: "v"(g), "v"(lds_byte_addr(lds)) : "memory");
#endif
}

__device__ __forceinline__ void wait_asynccnt0() {
#if __has_builtin(__builtin_amdgcn_s_wait_asynccnt)
    __builtin_amdgcn_s_wait_asynccnt(0);
#else
    asm volatile("s_wait_asynccnt 0" ::: "memory");
#endif
}

// One wave (32 lanes) gathers one 4 KB embedding row through its private 4 KB
// LDS slice: 8 async 128b loads per lane fill it, one s_wait_asynccnt,
// 8 async 128b stores drain it. 8 waves per block, no barriers needed.
__global__ __launch_bounds__(256) void EmbeddingMul_gather_async(
    const int* __restrict__ indices,   // (rows) flattened (bptt, bsize)
    const float* __restrict__ weight,  // (depth, em)
    float* __restrict__ out,           // (rows, em)
    int rows, int em, int depth) {
    __shared__ __align__(16) char lds[8 * 4096];

    const int wid  = (int)(threadIdx.x >> 5);
    const int lane = (int)(threadIdx.x & 31);
    const int row  = (int)blockIdx.x * 8 + wid;
    if (row >= rows) return;  // wave-uniform exit

    unsigned tok = (unsigned)indices[row];   // wave-uniform
    if (tok >= (unsigned)depth) tok = 0;     // defensive clamp; inputs in-range

    const char* src   = (const char*)weight + (unsigned long long)tok * (unsigned long long)em * 4ull;
    char*       dst   = (char*)out          + (unsigned long long)row * (unsigned long long)em * 4ull;
    char*       myLds = lds + wid * 4096;

    const int base   = lane * 16;                 // 16 B per lane per chunk
    int nChunk = (em * 4) >> 9;                   // 512 B per wave per chunk
    if (nChunk > 8) nChunk = 8;                   // LDS slice is 4 KB (em == 1024)

    #pragma unroll 8
    for (int i = 0; i < nChunk; ++i)
        async_g2lds_b128(myLds + base + i * 512, src + base + i * 512);

    wait_asynccnt0();  // LDS slice filled; safe for async stores to read it

    #pragma unroll 8
    for (int i = 0; i < nChunk; ++i)
        async_lds2g_b128(dst + base + i * 512, myLds + base + i * 512);
    // S_ENDPGM performs an implicit wait-idle, draining remaining ASYNCcnt.
}

extern "C" void kernel_launch(void* const* d_in, const int* in_sizes, int n_in,
                              void* d_out, int out_size, void* d_ws, size_t ws_size,
                              hipStream_t stream) {
    const int*   indices = (const int*)d_in[0];    // (64,32) int32
    const float* weight  = (const float*)d_in[1];  // (32000,1024) fp32
    float*       out     = (float*)d_out;          // (64,32,1024) fp32

    const int rows  = in_sizes[0];                 // 2048
    const int em    = out_size / rows;             // 1024
    const int depth = in_sizes[1] / em;            // 32000

    const int blocks = (rows + 7) / 8;             // 8 rows (waves) per block
    EmbeddingMul_gather_async<<<blocks, 256, 0, stream>>>(
        indices, weight, out, rows, em, depth);
}